// GCN_21320217658153
// MI455X (gfx1250) — compile-verified
//
#include <hip/hip_runtime.h>

#define N_NODES 100000
#define N_EDGES 3200000
#define E_TOT   (N_EDGES + N_NODES)
#define IN_C    128
#define HID_C   16
#define OUT_C   5
#define NEG_SLOPE 0.2f

typedef __attribute__((ext_vector_type(2))) float v2f;
typedef __attribute__((ext_vector_type(8))) float v8f;

// ---------- helpers ----------

// Monotone float<->signed-int key (involution). Lets us use integer atomicMax
// for the per-destination segment max of edge scores.
__device__ __forceinline__ int f2key(float f) {
  int k = __float_as_int(f);
  return (k >= 0) ? k : (k ^ 0x7fffffff);
}
__device__ __forceinline__ float key2f(int k) {
  return __int_as_float((k >= 0) ? k : (k ^ 0x7fffffff));
}

__device__ __forceinline__ float lrelu(float e) {
  return (e > 0.0f) ? e : NEG_SLOPE * e;
}

// Edge i in [0, E_TOT): first N_EDGES from edge_index (flat (2,E): row0=src,
// row1=dst), remainder are the self-loops appended by the reference.
__device__ __forceinline__ void edge_sd(const int* __restrict__ ei, int i,
                                        int& s, int& d) {
  if (i < N_EDGES) { s = ei[i]; d = ei[N_EDGES + i]; }
  else             { s = d = i - N_EDGES; }
}

// ---------- kernel 1: h1 = x @ W1 via V_WMMA_F32_16X16X4_F32 ----------
// One wave per 16-row tile; K=128 consumed as 32 chained k=4 WMMAs.
// A (16x4 f32) layout: lanes 0-15 -> M=lane, {K0,K1}; lanes 16-31 -> {K2,K3}.
// B (4x16 f32) layout mirrors A with N across lanes.
// C/D (16x16 f32): VGPR r, lane l -> row r + (l>=16 ? 8 : 0), col l&15.
__global__ __launch_bounds__(128)
void gemm1_wmma(const float* __restrict__ x, const float* __restrict__ W1,
                float* __restrict__ h1) {
  const int wave = blockIdx.x * (blockDim.x >> 5) + (threadIdx.x >> 5);
  const int row0 = wave * 16;
  if (row0 >= N_NODES) return;          // uniform per wave -> EXEC all-1s
  const int lane  = threadIdx.x & 31;
  const int m     = lane & 15;
  const int khalf = (lane >> 4) << 1;   // 0 or 2

  const float* __restrict__ xrow = x + (size_t)(row0 + m) * IN_C;

  v8f c = {0.f, 0.f, 0.f, 0.f, 0.f, 0.f, 0.f, 0.f};
#pragma unroll 8
  for (int k = 0; k < IN_C; k += 4) {
    v2f a, b;
    a.x = xrow[k + khalf];
    a.y = xrow[k + khalf + 1];
    b.x = W1[(k + khalf)     * HID_C + m];
    b.y = W1[(k + khalf + 1) * HID_C + m];
    c = __builtin_amdgcn_wmma_f32_16x16x4_f32(
        /*neg_a=*/false, a, /*neg_b=*/false, b,
        /*c_mod=*/(short)0, c, /*reuse_a=*/false, /*reuse_b=*/false);
  }

  const int rowoff = (lane >> 4) << 3;  // 0 or 8
#pragma unroll
  for (int r = 0; r < 8; ++r)
    h1[(size_t)(row0 + r + rowoff) * HID_C + m] = c[r];
}

// ---------- per-node attention coefficients: a_src/a_dst = h . att ----------
template <int C>
__global__ void attn_node(const float* __restrict__ h,
                          const float* __restrict__ att_s,
                          const float* __restrict__ att_d,
                          float* __restrict__ a_s, float* __restrict__ a_d) {
  int n = blockIdx.x * blockDim.x + threadIdx.x;
  if (n >= N_NODES) return;
  float s = 0.f, d = 0.f;
#pragma unroll
  for (int c = 0; c < C; ++c) {
    float v = h[(size_t)n * C + c];
    s += v * att_s[c];
    d += v * att_d[c];
  }
  a_s[n] = s;
  a_d[n] = d;
}

// ---------- per-layer accumulator init (must run every call) ----------
__global__ void init_layer(int* __restrict__ mkey, float* __restrict__ denom,
                           float* __restrict__ agg, int aggN) {
  int i = blockIdx.x * blockDim.x + threadIdx.x;
  if (i < N_NODES) { mkey[i] = (int)0x80000000; denom[i] = 0.0f; }
  if (i < aggN)    { agg[i] = 0.0f; }
}

// ---------- edge pass 1: segment max of leaky-relu scores ----------
__global__ void edge_max_k(const int* __restrict__ ei,
                           const float* __restrict__ a_s,
                           const float* __restrict__ a_d,
                           int* __restrict__ mkey) {
  int i = blockIdx.x * blockDim.x + threadIdx.x;
  if (i >= E_TOT) return;
  int s, d;
  edge_sd(ei, i, s, d);
  float e = lrelu(a_s[s] + a_d[d]);
  atomicMax(&mkey[d], f2key(e));
}

// ---------- edge pass 2: denom = segment_sum(exp(e - m)) ----------
__global__ void edge_expsum_k(const int* __restrict__ ei,
                              const float* __restrict__ a_s,
                              const float* __restrict__ a_d,
                              const int* __restrict__ mkey,
                              float* __restrict__ denom) {
  int i = blockIdx.x * blockDim.x + threadIdx.x;
  if (i >= E_TOT) return;
  int s, d;
  edge_sd(ei, i, s, d);
  float e = lrelu(a_s[s] + a_d[d]);
  atomicAdd(&denom[d], __expf(e - key2f(mkey[d])));
}

// ---------- edge pass 3: agg[dst] += alpha * h[src] ----------
template <int C>
__global__ void edge_agg_k(const int* __restrict__ ei,
                           const float* __restrict__ a_s,
                           const float* __restrict__ a_d,
                           const int* __restrict__ mkey,
                           const float* __restrict__ denom,
                           const float* __restrict__ h,
                           float* __restrict__ agg) {
  int i = blockIdx.x * blockDim.x + threadIdx.x;
  if (i >= E_TOT) return;
  int s, d;
  edge_sd(ei, i, s, d);
  float e  = lrelu(a_s[s] + a_d[d]);
  float ex = __expf(e - key2f(mkey[d]));
  float alpha = ex / (denom[d] + 1e-16f);
  const float* __restrict__ hs = h + (size_t)s * C;
  float* __restrict__ od = agg + (size_t)d * C;
#pragma unroll
  for (int c = 0; c < C; ++c) atomicAdd(&od[c], hs[c] * alpha);
}

// ---------- h1r = relu(agg1 + b1), written back over h1 ----------
__global__ void finish1_k(const float* __restrict__ agg1,
                          const float* __restrict__ b1,
                          float* __restrict__ h1r) {
  int i = blockIdx.x * blockDim.x + threadIdx.x;
  if (i >= N_NODES * HID_C) return;
  h1r[i] = fmaxf(agg1[i] + b1[i % HID_C], 0.0f);
}

// ---------- layer-2 projection (16x5, scalar) + attention coeffs ----------
__global__ void gemm2_k(const float* __restrict__ hin,
                        const float* __restrict__ W2,
                        const float* __restrict__ as2,
                        const float* __restrict__ ad2,
                        float* __restrict__ h2,
                        float* __restrict__ a_s, float* __restrict__ a_d) {
  int n = blockIdx.x * blockDim.x + threadIdx.x;
  if (n >= N_NODES) return;
  float acc[OUT_C] = {0.f, 0.f, 0.f, 0.f, 0.f};
#pragma unroll
  for (int k = 0; k < HID_C; ++k) {
    float v = hin[(size_t)n * HID_C + k];
#pragma unroll
    for (int j = 0; j < OUT_C; ++j) acc[j] += v * W2[k * OUT_C + j];
  }
  float s = 0.f, d = 0.f;
#pragma unroll
  for (int j = 0; j < OUT_C; ++j) {
    h2[(size_t)n * OUT_C + j] = acc[j];
    s += acc[j] * as2[j];
    d += acc[j] * ad2[j];
  }
  a_s[n] = s;
  a_d[n] = d;
}

// ---------- final: out = log_softmax(agg2 + b2) ----------
__global__ void logsoftmax_k(const float* __restrict__ agg2,
                             const float* __restrict__ b2,
                             float* __restrict__ out) {
  int n = blockIdx.x * blockDim.x + threadIdx.x;
  if (n >= N_NODES) return;
  float v[OUT_C];
  float mx = -3.402823466e38f;
#pragma unroll
  for (int c = 0; c < OUT_C; ++c) {
    v[c] = agg2[(size_t)n * OUT_C + c] + b2[c];
    mx = fmaxf(mx, v[c]);
  }
  float se = 0.f;
#pragma unroll
  for (int c = 0; c < OUT_C; ++c) se += __expf(v[c] - mx);
  float l = logf(se) + mx;
#pragma unroll
  for (int c = 0; c < OUT_C; ++c)
    out[(size_t)n * OUT_C + c] = v[c] - l;
}

// ---------- launcher ----------
extern "C" void kernel_launch(void* const* d_in, const int* in_sizes, int n_in,
                              void* d_out, int out_size, void* d_ws,
                              size_t ws_size, hipStream_t stream) {
  (void)in_sizes; (void)n_in; (void)out_size; (void)ws_size;

  const float* x   = (const float*)d_in[0];
  const int*   ei  = (const int*)d_in[1];
  const float* W1  = (const float*)d_in[2];
  const float* as1 = (const float*)d_in[3];
  const float* ad1 = (const float*)d_in[4];
  const float* b1  = (const float*)d_in[5];
  const float* W2  = (const float*)d_in[6];
  const float* as2 = (const float*)d_in[7];
  const float* ad2 = (const float*)d_in[8];
  const float* b2  = (const float*)d_in[9];
  float* out = (float*)d_out;

  // Workspace carve-up (all 16B aligned sizes).
  char* ws = (char*)d_ws;
  float* h1   = (float*)ws; ws += (size_t)N_NODES * HID_C * sizeof(float);
  float* agg1 = (float*)ws; ws += (size_t)N_NODES * HID_C * sizeof(float);
  float* h2   = (float*)ws; ws += (size_t)N_NODES * OUT_C * sizeof(float);
  float* agg2 = (float*)ws; ws += (size_t)N_NODES * OUT_C * sizeof(float);
  float* a_s  = (float*)ws; ws += (size_t)N_NODES * sizeof(float);
  float* a_d  = (float*)ws; ws += (size_t)N_NODES * sizeof(float);
  int*   mkey = (int*)ws;   ws += (size_t)N_NODES * sizeof(int);
  float* den  = (float*)ws; ws += (size_t)N_NODES * sizeof(float);

  const int B = 256;
  const int gNodes = (N_NODES + B - 1) / B;
  const int gEdges = (E_TOT + B - 1) / B;
  const int gAgg1  = (N_NODES * HID_C + B - 1) / B;
  const int gAgg2  = (N_NODES * OUT_C + B - 1) / B;
  // GEMM1: 4 waves / block, 1 wave per 16-row tile, 6250 tiles.
  const int wavesNeeded = (N_NODES + 15) / 16;
  const int gGemm1 = (wavesNeeded + 3) / 4;

  // ---- layer 1 ----
  gemm1_wmma<<<gGemm1, 128, 0, stream>>>(x, W1, h1);
  attn_node<HID_C><<<gNodes, B, 0, stream>>>(h1, as1, ad1, a_s, a_d);
  init_layer<<<gAgg1, B, 0, stream>>>(mkey, den, agg1, N_NODES * HID_C);
  edge_max_k<<<gEdges, B, 0, stream>>>(ei, a_s, a_d, mkey);
  edge_expsum_k<<<gEdges, B, 0, stream>>>(ei, a_s, a_d, mkey, den);
  edge_agg_k<HID_C><<<gEdges, B, 0, stream>>>(ei, a_s, a_d, mkey, den, h1, agg1);
  finish1_k<<<gAgg1, B, 0, stream>>>(agg1, b1, h1);   // h1 := relu(agg1+b1)

  // ---- layer 2 ----
  gemm2_k<<<gNodes, B, 0, stream>>>(h1, W2, as2, ad2, h2, a_s, a_d);
  init_layer<<<gAgg2, B, 0, stream>>>(mkey, den, agg2, N_NODES * OUT_C);
  edge_max_k<<<gEdges, B, 0, stream>>>(ei, a_s, a_d, mkey);
  edge_expsum_k<<<gEdges, B, 0, stream>>>(ei, a_s, a_d, mkey, den);
  edge_agg_k<OUT_C><<<gEdges, B, 0, stream>>>(ei, a_s, a_d, mkey, den, h2, agg2);
  logsoftmax_k<<<gNodes, B, 0, stream>>>(agg2, b2, out);
}